// intopt_74569222193428
// MI455X (gfx1250) — compile-verified
//
#include <hip/hip_runtime.h>

#define N_DIM 300
#define M_DIM 100
#define MPAD 112
#define MS   116
#define WS   18
#define NIT  40
#define DAMP 1e-5f
#define SIG  0.1f
#define STEPF 0.9f
#define NTHREADS 256
#define NWAVE (NTHREADS / 32)

typedef __attribute__((ext_vector_type(2))) float v2f;
typedef __attribute__((ext_vector_type(8))) float v8f;

__device__ __forceinline__ void breduce_sum(float v, float* red, float* out, int tid) {
  __syncthreads();
#pragma unroll
  for (int o = 16; o > 0; o >>= 1) v += __shfl_down(v, o, 32);
  if ((tid & 31) == 0) red[tid >> 5] = v;
  __syncthreads();
  if (tid == 0) {
    float s = 0.f;
    for (int i = 0; i < NWAVE; ++i) s += red[i];
    *out = s;
  }
  __syncthreads();
}

__device__ __forceinline__ void breduce_min(float v, float* red, float* out, int tid) {
  __syncthreads();
#pragma unroll
  for (int o = 16; o > 0; o >>= 1) v = fminf(v, __shfl_down(v, o, 32));
  if ((tid & 31) == 0) red[tid >> 5] = v;
  __syncthreads();
  if (tid == 0) {
    float s = red[0];
    for (int i = 1; i < NWAVE; ++i) s = fminf(s, red[i]);
    *out = s;
  }
  __syncthreads();
}

__global__ __launch_bounds__(NTHREADS, 1)
void ipm_kernel(const float* __restrict__ Cg, const float* __restrict__ Ag,
                const float* __restrict__ bg, float* __restrict__ Xout) {
  __shared__ float S[M_DIM * N_DIM];      // scaled constraint matrix S = A*diag(s)
  __shared__ float Mm[MPAD * MS];         // normal matrix -> LDL^T factor (V = L*D, diag = D)
  __shared__ float wbuf[MPAD * WS];       // W = V * D^{-1/2} panel for WMMA trailing update
  __shared__ float xv[N_DIM], zv[N_DIM], sv[N_DIM], siv[N_DIM];
  __shared__ float rdv[N_DIM], tv[N_DIM], uv[N_DIM];
  __shared__ float yv[M_DIM], rhs[MPAD], dinvv[MPAD];
  __shared__ float red[NWAVE];
  __shared__ float scal[4];

  const int tid = threadIdx.x;
  const int b = blockIdx.x;
  const float* cb = Cg + (size_t)b * N_DIM;

  for (int j = tid; j < N_DIM; j += NTHREADS) { xv[j] = 1.f; zv[j] = 1.f; }
  for (int i = tid; i < MPAD; i += NTHREADS) {
    if (i < M_DIM) yv[i] = 0.f;
    rhs[i] = 0.f;
  }
  __syncthreads();

  for (int it = 0; it < NIT; ++it) {
    // ---- A: s = sqrt(x/z), 1/s, x/s = sqrt(xz); mu = mean(x*z)
    float part = 0.f;
    for (int k = tid; k < N_DIM; k += NTHREADS) {
      float xk = xv[k], zk = zv[k];
      float xz = xk * zk;
      part += xz;
      float sk = sqrtf(xk / zk);
      sv[k] = sk;
      siv[k] = 1.f / sk;
      tv[k] = sqrtf(xz);
    }
    breduce_sum(part, red, &scal[0], tid);
    const float mu = scal[0] * (1.f / (float)N_DIM);

    // ---- B: S = A * diag(s)
    for (int idx = tid; idx < M_DIM * N_DIM; idx += NTHREADS) {
      int k = idx % N_DIM;
      S[idx] = Ag[idx] * sv[k];
    }
    __syncthreads();

    // ---- C: rp = b - S (x/s) -> rhs ; rd = c - diag(1/s) S^T y - z
    for (int i = tid; i < M_DIM; i += NTHREADS) {
      const float* Si = &S[i * N_DIM];
      float acc = 0.f;
      for (int k = 0; k < N_DIM; ++k) acc += Si[k] * tv[k];
      rhs[i] = bg[i] - acc;
    }
    for (int j = tid; j < N_DIM; j += NTHREADS) {
      float acc = 0.f;
      for (int i = 0; i < M_DIM; ++i) acc += S[i * N_DIM + j] * yv[i];
      rdv[j] = cb[j] - siv[j] * acc - zv[j];
    }
    __syncthreads();

    // ---- D: t' = s*(rd - rc/x); rhs += S t'
    for (int k = tid; k < N_DIM; k += NTHREADS) {
      float xk = xv[k];
      float rck = SIG * mu - xk * zv[k];
      tv[k] = sv[k] * (rdv[k] - rck / xk);
    }
    __syncthreads();
    for (int i = tid; i < M_DIM; i += NTHREADS) {
      const float* Si = &S[i * N_DIM];
      float acc = 0.f;
      for (int k = 0; k < N_DIM; ++k) acc += Si[k] * tv[k];
      rhs[i] += acc;
    }

    // ---- E: M = S S^T (+damping), lower-triangle tiles only (28 of 49), WMMA f32 16x16x4
    {
      const int lane = tid & 31, wid = tid >> 5;
      const int half = lane >> 4;
      for (int t = wid; t < 28; t += NWAVE) {
        int ti = 0, rem = t;
        while (rem >= ti + 1) { rem -= ti + 1; ++ti; }
        const int I = ti, J = rem;                      // I >= J
        int rA = I * 16 + (lane & 15); if (rA >= M_DIM) rA = M_DIM - 1;
        int rB = J * 16 + (lane & 15); if (rB >= M_DIM) rB = M_DIM - 1;
        const float* pA = &S[rA * N_DIM + half * 2];
        const float* pB = &S[rB * N_DIM + half * 2];
        v8f acc = {0.f, 0.f, 0.f, 0.f, 0.f, 0.f, 0.f, 0.f};
        for (int kk = 0; kk < N_DIM; kk += 4) {
          v2f av, bv;
          av.x = pA[kk]; av.y = pA[kk + 1];
          bv.x = pB[kk]; bv.y = pB[kk + 1];
          acc = __builtin_amdgcn_wmma_f32_16x16x4_f32(
              false, av, false, bv, (short)0, acc, false, false);
        }
        const int row0 = I * 16 + half * 8;
        const int col  = J * 16 + (lane & 15);
#pragma unroll
        for (int v = 0; v < 8; ++v) {
          float val = acc[v];
          int r = row0 + v;
          if (r == col) val += DAMP;
          Mm[r * MS + col] = val;
        }
      }
    }
    __syncthreads();
    // padding rows -> identity so all 7 blocks of the factorization are uniform 16x16
    for (int idx = tid; idx < 12 * MPAD; idx += NTHREADS) {
      int r = M_DIM + idx / MPAD, c2 = idx % MPAD;
      Mm[r * MS + c2] = (r == c2) ? 1.f : 0.f;
    }

    // ---- F: blocked LDL^T (NB=16). Mm[i][k] ends as V=L*D (lower), Mm[k][k] = d[k].
    for (int blk = 0; blk < 7; ++blk) {
      const int k0 = blk * 16;
      // F1: diagonal block, unblocked LDL (1 barrier per step; col k is read-only in its step)
      for (int kk = 0; kk < 15; ++kk) {
        __syncthreads();
        const int k = k0 + kk, tsz = 15 - kk;
        const float inv = 1.f / Mm[k * MS + k];
        for (int idx = tid; idx < tsz * tsz; idx += NTHREADS) {
          int i = k + 1 + idx / tsz, j = k + 1 + idx % tsz;
          Mm[i * MS + j] -= Mm[i * MS + k] * inv * Mm[j * MS + k];
        }
      }
      __syncthreads();
      // F2: panel solve V21 = A21 * L11^{-T}  (one row per thread, in registers)
      {
        const int r = k0 + 16 + tid;
        if (r < MPAD) {
          float a[16], invd[16];
#pragma unroll
          for (int c2 = 0; c2 < 16; ++c2) {
            a[c2] = Mm[r * MS + k0 + c2];
            invd[c2] = 1.f / Mm[(k0 + c2) * MS + (k0 + c2)];
          }
#pragma unroll
          for (int c2 = 1; c2 < 16; ++c2) {
            float val = a[c2];
#pragma unroll
            for (int t2 = 0; t2 < 15; ++t2) {
              if (t2 < c2) val -= a[t2] * invd[t2] * Mm[(k0 + c2) * MS + (k0 + t2)];
            }
            a[c2] = val;
          }
#pragma unroll
          for (int c2 = 0; c2 < 16; ++c2) Mm[r * MS + k0 + c2] = a[c2];
        }
      }
      __syncthreads();
      // F3: stage W = V * D^{-1/2}
      {
        const int nrows = MPAD - (k0 + 16);
        for (int idx = tid; idx < nrows * 16; idx += NTHREADS) {
          int r = k0 + 16 + (idx >> 4), c2 = idx & 15;
          float d = Mm[(k0 + c2) * MS + (k0 + c2)];
          wbuf[r * WS + c2] = Mm[r * MS + k0 + c2] * rsqrtf(d);
        }
      }
      __syncthreads();
      // F4: rank-16 trailing update A22 -= W W^T via WMMA (lower-triangle tiles)
      {
        const int lane = tid & 31, wid = tid >> 5;
        const int half = lane >> 4;
        const int T = 6 - blk;
        const int tcount = T * (T + 1) / 2;
        for (int t = wid; t < tcount; t += NWAVE) {
          int ti = 0, rem = t;
          while (rem >= ti + 1) { rem -= ti + 1; ++ti; }
          const int I = blk + 1 + ti, J = blk + 1 + rem;
          const int rA = I * 16 + (lane & 15);
          const int rB = J * 16 + (lane & 15);
          const float* pA = &wbuf[rA * WS + half * 2];
          const float* pB = &wbuf[rB * WS + half * 2];
          v8f acc = {0.f, 0.f, 0.f, 0.f, 0.f, 0.f, 0.f, 0.f};
#pragma unroll
          for (int kk = 0; kk < 16; kk += 4) {
            v2f av, bv;
            av.x = pA[kk]; av.y = pA[kk + 1];
            bv.x = pB[kk]; bv.y = pB[kk + 1];
            acc = __builtin_amdgcn_wmma_f32_16x16x4_f32(
                false, av, false, bv, (short)0, acc, false, false);
          }
          const int row0 = I * 16 + half * 8;
          const int col  = J * 16 + (lane & 15);
#pragma unroll
          for (int v = 0; v < 8; ++v) Mm[(row0 + v) * MS + col] -= acc[v];
        }
      }
      __syncthreads();
    }

    // ---- G: solves with L unit-lower (L[i][k] = V[i][k]/d[k]).  dy ends up in rhs.
    for (int i = tid; i < MPAD; i += NTHREADS) dinvv[i] = 1.f / Mm[i * MS + i];
    __syncthreads();
    for (int k = 0; k < M_DIM; ++k) {                 // L w = rhs (w[k]=rhs[k], unit diag)
      const float wk = rhs[k] * dinvv[k];
      for (int i = k + 1 + tid; i < M_DIM; i += NTHREADS) rhs[i] -= Mm[i * MS + k] * wk;
      __syncthreads();
    }
    for (int k = tid; k < M_DIM; k += NTHREADS) rhs[k] *= dinvv[k];  // u = D^{-1} w
    __syncthreads();
    for (int k = M_DIM - 1; k >= 0; --k) {            // L^T dy = u
      const float dk = rhs[k];
      for (int i = tid; i < k; i += NTHREADS) rhs[i] -= Mm[k * MS + i] * dinvv[i] * dk;
      __syncthreads();
    }

    // ---- H: dx, dz, fraction-to-boundary (dy == rhs)
    float apl = 1e30f, adl = 1e30f;
    for (int j = tid; j < N_DIM; j += NTHREADS) {
      float acc = 0.f;
      for (int i = 0; i < M_DIM; ++i) acc += S[i * N_DIM + j] * rhs[i];
      float uj = siv[j] * acc;
      float xj = xv[j], zj = zv[j];
      float rcx = (SIG * mu - xj * zj) / xj;
      float sj = sv[j];
      float dxj = sj * sj * (uj - rdv[j] + rcx);
      float dzj = rdv[j] - uj;
      tv[j] = dxj; uv[j] = dzj;
      if (dxj < 0.f) apl = fminf(apl, -xj / dxj);
      if (dzj < 0.f) adl = fminf(adl, -zj / dzj);
    }
    breduce_min(apl, red, &scal[2], tid);
    breduce_min(adl, red, &scal[3], tid);
    if (tid == 0) scal[1] = fminf(STEPF * fminf(scal[2], scal[3]), 1.f);
    __syncthreads();
    const float alpha = scal[1];
    for (int j = tid; j < N_DIM; j += NTHREADS) { xv[j] += alpha * tv[j]; zv[j] += alpha * uv[j]; }
    for (int i = tid; i < M_DIM; i += NTHREADS) yv[i] += alpha * rhs[i];
    __syncthreads();
  }

  for (int j = tid; j < N_DIM; j += NTHREADS) Xout[(size_t)b * N_DIM + j] = xv[j];
}

extern "C" void kernel_launch(void* const* d_in, const int* in_sizes, int n_in,
                              void* d_out, int out_size, void* d_ws, size_t ws_size,
                              hipStream_t stream) {
  const float* c = (const float*)d_in[0];
  const float* A = (const float*)d_in[1];
  const float* b = (const float*)d_in[2];
  float* out = (float*)d_out;
  (void)n_in; (void)out_size; (void)d_ws; (void)ws_size;
  int B = in_sizes[0] / N_DIM;
  ipm_kernel<<<B, NTHREADS, 0, stream>>>(c, A, b, out);
}